// ISDALoss_893353198279
// MI455X (gfx1250) — compile-verified
//
#include <hip/hip_runtime.h>
#include <hip/hip_bf16.h>
#include <math.h>

// Problem constants (from reference): N=8192 rows, C=751 classes, A=2048 feats.
#define FDIM 2048          // A
#define K2   4096          // fused K: [CV | CV*W] x [W^2 | -2W]
#define NCLS 751           // C
#define CP   768           // C padded to multiple of 32 (48 x 16 tiles)
#define NROW 8192          // N
#define KC   128           // K-chunk staged in LDS
#define LDSW 132           // LDS row stride in floats (132*4B = 528B = 33*16B: float4-aligned, bank-conflict-free)

typedef __attribute__((ext_vector_type(2))) float v2f;
typedef __attribute__((ext_vector_type(8))) float v8f;

// ---------------------------------------------------------------------------
// Kernel 1: M[l,c] = sum_a CV[l,a]*W[c,a]^2 - 2*sum_a CV[l,a]*W[l,a]*W[c,a]
// Single fp32 WMMA GEMM, K=4096 (first 2048 = term1 part, second 2048 = -2*term2).
// WG = 128 threads (4 waves as 2x2), 32x32 output tile, LDS-staged K-chunks.
// ---------------------------------------------------------------------------
__global__ __launch_bounds__(128)
void isda_class_gemm(const float* __restrict__ W, const float* __restrict__ CV,
                     float* __restrict__ Mout) {
    __shared__ float As[32][LDSW];   // A' rows (l classes)
    __shared__ float Bs[32][LDSW];   // B' rows (c classes)

    const int t    = threadIdx.x;
    const int lane = t & 31;
    const int wave = t >> 5;
    const int wm   = wave >> 1;      // wave row in 2x2
    const int wn   = wave & 1;       // wave col in 2x2
    const int h    = lane >> 4;      // half-wave
    const int lid  = lane & 15;

    const int mBase = blockIdx.y * 32;   // l-class base
    const int nBase = blockIdx.x * 32;   // c-class base

    // staging decomposition: 32 rows x 32 float4-cols per matrix
    const int c4 = t & 31;           // float4 column within chunk
    const int r0 = t >> 5;           // 0..3 -> rows r0*8 .. r0*8+7

    v8f acc = {0.f, 0.f, 0.f, 0.f, 0.f, 0.f, 0.f, 0.f};

    for (int k0 = 0; k0 < K2; k0 += KC) {
        const bool phase2 = (k0 >= FDIM);              // uniform per chunk (KC | FDIM)
        const int  kk0    = phase2 ? (k0 - FDIM) : k0; // column into W / CV
        const int  col    = kk0 + c4 * 4;

        // ---- stage A' and B' chunk into LDS (computed on the fly) ----
        #pragma unroll
        for (int rr = 0; rr < 8; ++rr) {
            const int r = r0 * 8 + rr;
            // A' row: CV[l,k]            (phase 1)
            //         CV[l,k]*W[l,k]     (phase 2)
            {
                const int gr = mBase + r;
                float4 v = make_float4(0.f, 0.f, 0.f, 0.f);
                if (gr < NCLS) {
                    const float4 cv = *(const float4*)(CV + (size_t)gr * FDIM + col);
                    if (!phase2) {
                        v = cv;
                    } else {
                        const float4 w = *(const float4*)(W + (size_t)gr * FDIM + col);
                        v = make_float4(cv.x * w.x, cv.y * w.y, cv.z * w.z, cv.w * w.w);
                    }
                }
                *(float4*)&As[r][c4 * 4] = v;
            }
            // B' row: W[c,k]^2           (phase 1)
            //         -2*W[c,k]          (phase 2)
            {
                const int gc = nBase + r;
                float4 v = make_float4(0.f, 0.f, 0.f, 0.f);
                if (gc < NCLS) {
                    const float4 w = *(const float4*)(W + (size_t)gc * FDIM + col);
                    if (!phase2) {
                        v = make_float4(w.x * w.x, w.y * w.y, w.z * w.z, w.w * w.w);
                    } else {
                        v = make_float4(-2.f * w.x, -2.f * w.y, -2.f * w.z, -2.f * w.w);
                    }
                }
                *(float4*)&Bs[r][c4 * 4] = v;
            }
        }
        __syncthreads();

        // ---- compute: 32 x V_WMMA_F32_16X16X4_F32 per chunk per wave ----
        // A operand layout (16x4 f32): lane(h,lid) holds A[M=lid, K=2h],[K=2h+1] -> contiguous float2
        // B operand layout (4x16 f32): lane(h,lid) holds B[K=2h, N=lid],[K=2h+1] -> contiguous float2
        const float* arow = &As[wm * 16 + lid][2 * h];
        const float* brow = &Bs[wn * 16 + lid][2 * h];
        #pragma unroll
        for (int kk = 0; kk < KC; kk += 4) {
            v2f a = *(const v2f*)(arow + kk);
            v2f b = *(const v2f*)(brow + kk);
            acc = __builtin_amdgcn_wmma_f32_16x16x4_f32(
                /*neg_a=*/false, a, /*neg_b=*/false, b,
                /*c_mod=*/(short)0, acc, /*reuse_a=*/false, /*reuse_b=*/false);
        }
        __syncthreads();
    }

    // ---- write D: lane(h,lid), VGPR v -> element (M = v + 8h, N = lid) ----
    const int colOut = nBase + wn * 16 + lid;
    #pragma unroll
    for (int v = 0; v < 8; ++v) {
        const int rowOut = mBase + wm * 16 + v + 8 * h;
        Mout[(size_t)rowOut * CP + colOut] = acc[v];
    }
}

// ---------------------------------------------------------------------------
// Kernel 2: per-row cross-entropy on aug[n,c] = y[n,c] + 0.5*ratio*(M[l,c]-M[l,l])
// One 256-thread block per row; deterministic LDS tree reductions.
// Note aug[n, l] == y[n, l] exactly (sigma term vanishes at c == l).
// ---------------------------------------------------------------------------
__global__ __launch_bounds__(256)
void isda_row_loss(const float* __restrict__ y, const int* __restrict__ labels,
                   const float* __restrict__ Mmat, const float* __restrict__ ratio_p,
                   float* __restrict__ losses) {
    __shared__ float red[256];
    const int n = blockIdx.x;
    const int t = threadIdx.x;
    const int l = labels[n];
    const float scale = 0.5f * ratio_p[0];
    const float* yr = y + (size_t)n * NCLS;
    const float* Mr = Mmat + (size_t)l * CP;
    const float Mll = Mr[l];

    float vals[3];
    float vmax = -__builtin_inff();
    #pragma unroll
    for (int i = 0; i < 3; ++i) {
        const int c = t + i * 256;
        float a = -__builtin_inff();
        if (c < NCLS) a = yr[c] + scale * (Mr[c] - Mll);
        vals[i] = a;
        vmax = fmaxf(vmax, a);
    }
    red[t] = vmax;
    __syncthreads();
    #pragma unroll
    for (int s = 128; s > 0; s >>= 1) {
        if (t < s) red[t] = fmaxf(red[t], red[t + s]);
        __syncthreads();
    }
    const float m = red[0];
    __syncthreads();

    float sum = 0.f;
    #pragma unroll
    for (int i = 0; i < 3; ++i) {
        const int c = t + i * 256;
        if (c < NCLS) sum += expf(vals[i] - m);
    }
    red[t] = sum;
    __syncthreads();
    #pragma unroll
    for (int s = 128; s > 0; s >>= 1) {
        if (t < s) red[t] += red[t + s];
        __syncthreads();
    }
    if (t == 0) {
        // nll = logsumexp(aug) - aug[label];  aug[label] == y[n,l]
        losses[n] = logf(red[0]) + m - yr[l];
    }
}

// ---------------------------------------------------------------------------
// Kernel 3: mean over N row losses (single block, deterministic order).
// ---------------------------------------------------------------------------
__global__ __launch_bounds__(256)
void isda_mean(const float* __restrict__ losses, float* __restrict__ out) {
    __shared__ float red[256];
    float s = 0.f;
    for (int i = threadIdx.x; i < NROW; i += 256) s += losses[i];
    red[threadIdx.x] = s;
    __syncthreads();
    #pragma unroll
    for (int st = 128; st > 0; st >>= 1) {
        if (threadIdx.x < st) red[threadIdx.x] += red[threadIdx.x + st];
        __syncthreads();
    }
    if (threadIdx.x == 0) out[0] = red[0] / (float)NROW;
}

// ---------------------------------------------------------------------------
// Launch: inputs in setup_inputs() order:
//   0 features (unused by the reference math), 1 y, 2 labels, 3 weight,
//   4 CoVariance, 5 ratio
// ---------------------------------------------------------------------------
extern "C" void kernel_launch(void* const* d_in, const int* in_sizes, int n_in,
                              void* d_out, int out_size, void* d_ws, size_t ws_size,
                              hipStream_t stream) {
    (void)in_sizes; (void)n_in; (void)out_size; (void)ws_size;
    const float* y      = (const float*)d_in[1];
    const int*   labels = (const int*)d_in[2];
    const float* W      = (const float*)d_in[3];
    const float* CV     = (const float*)d_in[4];
    const float* ratio  = (const float*)d_in[5];

    float* Mmat   = (float*)d_ws;                                   // CP*CP floats (2.36 MB)
    float* losses = (float*)((char*)d_ws + (size_t)CP * CP * sizeof(float)); // NROW floats

    dim3 ggrid(CP / 32, CP / 32);   // 24 x 24 workgroups
    isda_class_gemm<<<ggrid, 128, 0, stream>>>(W, CV, Mmat);
    isda_row_loss<<<NROW, 256, 0, stream>>>(y, labels, Mmat, ratio, losses);
    isda_mean<<<1, 256, 0, stream>>>(losses, (float*)d_out);
}